// BoundaryLoss_7576322310173
// MI455X (gfx1250) — compile-verified
//
#include <hip/hip_runtime.h>

#define SMOOTH_F 1e-5f
#define SMOOTH_D 1e-5

#define WDIM   192
#define SLICE  (WDIM * WDIM)          // 36864 floats per z-slice
#define RY     8                      // output rows per tile
#define HY     (RY + 2)               // rows incl. y-halo
#define LZ     8                      // z-slices per chunk
#define NYT    (WDIM / RY)            // 24 y-tiles
#define NZC    (WDIM / LZ)            // 24 z-chunks
#define TILE_F (HY * WDIM)            // 1920 floats per LDS slot
#define NSLOT  4                      // ring depth (z-1..z+2 live concurrently)
#define SMEM_BYTES (2 * NSLOT * TILE_F * 4)   // 61440 B

// ---------------------------------------------------------------------------
// LDS byte address of a generic pointer to shared memory.
// ISA aperture rule: low 32 bits of a flat shared address are the LDS offset.
// ---------------------------------------------------------------------------
__device__ __forceinline__ unsigned lds_off(const void* p)
{
    return (unsigned)(unsigned long long)p;
}

// ---------------------------------------------------------------------------
// Asynchronously stage one z-slice row-tile (rows [ylo..ylo+nrows-1], full x)
// into an LDS slot via GLOBAL_LOAD_ASYNC_TO_LDS_B128 (ASYNCcnt-tracked).
// Out-of-volume slices are zero-filled (zero padding semantics).
// ---------------------------------------------------------------------------
__device__ __forceinline__ void stage_slice(const float* __restrict__ g,
                                            float* __restrict__ slot,
                                            int s, int ylo, int nrows,
                                            int dstrow0)
{
    const int t = threadIdx.x;
    if (s < 0 || s >= WDIM) {
        for (int i = t; i < TILE_F; i += 256) slot[i] = 0.0f;
        return;
    }
    const unsigned long long gbase =
        (unsigned long long)(const void*)(g + (size_t)s * SLICE + (size_t)ylo * WDIM);
    float* dst0 = slot + dstrow0 * WDIM;
    const int nchunks = nrows * (WDIM / 4);        // 16B chunks
    for (int i = t; i < nchunks; i += 256) {
        const unsigned voff = (unsigned)i * 16u;               // global byte offset
        const unsigned ldsa = lds_off(dst0) + (unsigned)i * 16u; // LDS byte address
        asm volatile("global_load_async_to_lds_b128 %0, %1, %2"
                     :: "v"(ldsa), "v"(voff), "s"(gbase)
                     : "memory");
    }
}

// ---------------------------------------------------------------------------
// Boundary magnitude of one aligned float4 span, all neighbors from LDS.
// Halo rows/slices contain genuine zeros -> only x-edge selects needed.
// ---------------------------------------------------------------------------
__device__ __forceinline__ float4 bmag(const float* __restrict__ slotM,
                                       const float* __restrict__ slotC,
                                       const float* __restrict__ slotP,
                                       int r, int xo, int c4)
{
    const float* pc = slotC + (r + 1) * WDIM + xo;
    const float4 c  = *(const float4*)pc;
    const float left  = (c4 == 0)  ? 0.0f : pc[-1];
    const float right = (c4 == 47) ? 0.0f : pc[4];
    const float4 ym = *(const float4*)(slotC + r * WDIM + xo);
    const float4 yp = *(const float4*)(slotC + (r + 2) * WDIM + xo);
    const float4 zm = *(const float4*)(slotM + (r + 1) * WDIM + xo);
    const float4 zp = *(const float4*)(slotP + (r + 1) * WDIM + xo);

    const float gx0 = c.y - left, gx1 = c.z - c.x, gx2 = c.w - c.y, gx3 = right - c.z;
    const float gy0 = yp.x - ym.x, gy1 = yp.y - ym.y, gy2 = yp.z - ym.z, gy3 = yp.w - ym.w;
    const float gz0 = zp.x - zm.x, gz1 = zp.y - zm.y, gz2 = zp.z - zm.z, gz3 = zp.w - zm.w;

    float4 r4;
    r4.x = sqrtf(fmaf(gx0, gx0, fmaf(gy0, gy0, fmaf(gz0, gz0, SMOOTH_F))));
    r4.y = sqrtf(fmaf(gx1, gx1, fmaf(gy1, gy1, fmaf(gz1, gz1, SMOOTH_F))));
    r4.z = sqrtf(fmaf(gx2, gx2, fmaf(gy2, gy2, fmaf(gz2, gz2, SMOOTH_F))));
    r4.w = sqrtf(fmaf(gx3, gx3, fmaf(gy3, gy3, fmaf(gz3, gz3, SMOOTH_F))));
    return r4;
}

// ---------------------------------------------------------------------------
__global__ void bl_init_kernel(double* __restrict__ acc)
{
    if (threadIdx.x < 3) acc[threadIdx.x] = 0.0;
}

// ---------------------------------------------------------------------------
// Main kernel: async-LDS software-pipelined 3D boundary stencil + reduction.
// Block = (b, y-tile, z-chunk). Ring of 4 LDS slots per tensor.
// ---------------------------------------------------------------------------
__global__ __launch_bounds__(256) void bl_main_kernel(
    const float* __restrict__ pred,
    const float* __restrict__ target,
    double* __restrict__ acc)
{
    extern __shared__ float sm[];
    float* smP = sm;                       // [NSLOT][TILE_F]
    float* smT = sm + NSLOT * TILE_F;

    const int bid = blockIdx.x;
    const int zc  = bid % NZC;
    const int yt  = (bid / NZC) % NYT;
    const int b   = bid / (NZC * NYT);
    const int y0  = yt * RY;
    const int z0  = zc * LZ;

    const float* gp = pred   + (size_t)b * WDIM * SLICE;
    const float* gt = target + (size_t)b * WDIM * SLICE;

    // --- one-time zero of y-halo rows never written by staging ---
    if (y0 == 0) {
        for (int i = threadIdx.x; i < NSLOT * WDIM; i += 256) {
            const int slot = i / WDIM, xo = i % WDIM;
            smP[slot * TILE_F + xo] = 0.0f;
            smT[slot * TILE_F + xo] = 0.0f;
        }
    }
    if (y0 + RY == WDIM) {
        for (int i = threadIdx.x; i < NSLOT * WDIM; i += 256) {
            const int slot = i / WDIM, xo = i % WDIM;
            smP[slot * TILE_F + (HY - 1) * WDIM + xo] = 0.0f;
            smT[slot * TILE_F + (HY - 1) * WDIM + xo] = 0.0f;
        }
    }

    const int ylo     = (y0 == 0) ? 0 : y0 - 1;
    const int yhi     = (y0 + RY >= WDIM) ? WDIM - 1 : y0 + RY;
    const int nrows   = yhi - ylo + 1;
    const int dstrow0 = ylo - (y0 - 1);    // 0 or 1

    // --- pipeline fill: slices z0-1, z0, z0+1 -> slots 0,1,2 ---
    #pragma unroll
    for (int k = 0; k < 3; ++k) {
        const int s = z0 - 1 + k;
        stage_slice(gp, smP + k * TILE_F, s, ylo, nrows, dstrow0);
        stage_slice(gt, smT + k * TILE_F, s, ylo, nrows, dstrow0);
    }
    asm volatile("s_wait_asynccnt 0" ::: "memory");
    __syncthreads();

    float si = 0.0f, sp = 0.0f, st = 0.0f;

    for (int zi = 0; zi < LZ; ++zi) {
        // prefetch slice z0+zi+2 into slot (zi+3)&3 (safe: last read in previous iter, before its barrier)
        if (zi + 2 <= LZ) {
            const int s = z0 + zi + 2;
            stage_slice(gp, smP + ((zi + 3) & 3) * TILE_F, s, ylo, nrows, dstrow0);
            stage_slice(gt, smT + ((zi + 3) & 3) * TILE_F, s, ylo, nrows, dstrow0);
        }
        const float* Pm = smP + ((zi    ) & 3) * TILE_F;
        const float* Pc = smP + ((zi + 1) & 3) * TILE_F;
        const float* Pp = smP + ((zi + 2) & 3) * TILE_F;
        const float* Tm = smT + ((zi    ) & 3) * TILE_F;
        const float* Tc = smT + ((zi + 1) & 3) * TILE_F;
        const float* Tp = smT + ((zi + 2) & 3) * TILE_F;

        for (int j = threadIdx.x; j < RY * 48; j += 256) {
            const int r  = j / 48;
            const int c4 = j - r * 48;
            const int xo = c4 * 4;
            const float4 pb = bmag(Pm, Pc, Pp, r, xo, c4);
            const float4 tb = bmag(Tm, Tc, Tp, r, xo, c4);
            si = fmaf(pb.x, tb.x, fmaf(pb.y, tb.y,
                 fmaf(pb.z, tb.z, fmaf(pb.w, tb.w, si))));
            sp += (pb.x + pb.y) + (pb.z + pb.w);
            st += (tb.x + tb.y) + (tb.z + tb.w);
        }
        asm volatile("s_wait_asynccnt 0" ::: "memory");
        __syncthreads();
    }

    // --- wave32 butterfly + cross-wave LDS + f64 device atomics ---
    for (int off = 16; off > 0; off >>= 1) {
        si += __shfl_xor(si, off, 32);
        sp += __shfl_xor(sp, off, 32);
        st += __shfl_xor(st, off, 32);
    }
    __shared__ float lsi[8], lsp[8], lst[8];
    const int lane = threadIdx.x & 31;
    const int wave = threadIdx.x >> 5;
    if (lane == 0) { lsi[wave] = si; lsp[wave] = sp; lst[wave] = st; }
    __syncthreads();
    if (threadIdx.x == 0) {
        float a = 0.f, bb = 0.f, cc = 0.f;
        #pragma unroll
        for (int w = 0; w < 8; ++w) { a += lsi[w]; bb += lsp[w]; cc += lst[w]; }
        atomicAdd(&acc[0], (double)a);
        atomicAdd(&acc[1], (double)bb);
        atomicAdd(&acc[2], (double)cc);
    }
}

// ---------------------------------------------------------------------------
__global__ void bl_final_kernel(const double* __restrict__ acc,
                                float* __restrict__ out)
{
    if (threadIdx.x == 0 && blockIdx.x == 0) {
        const double dice = (2.0 * acc[0] + SMOOTH_D) /
                            (acc[1] + acc[2] + SMOOTH_D);
        out[0] = (float)(1.0 - dice);
    }
}

// ---------------------------------------------------------------------------
extern "C" void kernel_launch(void* const* d_in, const int* in_sizes, int n_in,
                              void* d_out, int out_size, void* d_ws, size_t ws_size,
                              hipStream_t stream)
{
    const float* pred   = (const float*)d_in[0];
    const float* target = (const float*)d_in[1];
    float*  out = (float*)d_out;
    double* acc = (double*)d_ws;          // 3 doubles of scratch

    const int nblocks = 2 * NYT * NZC;    // 1152 workgroups

    bl_init_kernel<<<1, 32, 0, stream>>>(acc);
    bl_main_kernel<<<nblocks, 256, SMEM_BYTES, stream>>>(pred, target, acc);
    bl_final_kernel<<<1, 32, 0, stream>>>(acc, out);
}